// CBOW_55009941127479
// MI455X (gfx1250) — compile-verified
//
#include <hip/hip_runtime.h>
#include <hip/hip_bf16.h>
#include <math.h>
#include <stdint.h>

// ---------------- problem constants (match reference) ----------------
#define VOC      100000
#define EDIM     128
#define CTX      10
#define NNEG     20
#define NEG_PAD  32          // pad 20 -> 32 (two 16-row WMMA tiles)

// ---------------- workspace layout (floats) ----------------
// ws[0   .. 127]          : vo_embed accumulator
// ws[128 .. 255]          : vi_embed accumulator
// ws[256 .. 256+32*128)   : neg_embed accumulator (padded to 32 rows)
#define WS_VO   0
#define WS_VI   128
#define WS_NEG  256
#define WS_N    (256 + NEG_PAD * EDIM)   // 4352 floats

// slab of U rows staged in LDS per iteration
#define ROWS        64
#define U_STRIDE    132      // pad 128 -> 132: kills 2-way bank conflicts on B fetch
#define A_STRIDE    65       // pad 64 -> 65 : kills 16-way bank conflicts on A fetch
#define NSLAB       ((VOC + ROWS - 1) / ROWS)   // 1563

typedef __attribute__((ext_vector_type(2))) float v2f;
typedef __attribute__((ext_vector_type(8))) float v8f;

__device__ __forceinline__ void atomAddF(float* p, float v) {
    __hip_atomic_fetch_add(p, v, __ATOMIC_RELAXED, __HIP_MEMORY_SCOPE_AGENT);
}

__device__ __forceinline__ void waitAsync0() {
#if defined(__has_builtin) && __has_builtin(__builtin_amdgcn_s_wait_asynccnt)
    __builtin_amdgcn_s_wait_asynccnt(0);
#else
    asm volatile("s_wait_asynccnt 0x0" ::: "memory");
#endif
}

// CDNA5 async copy: 16B per lane, global -> LDS, no VGPR round-trip (ASYNCcnt).
// ISA 10.2: generic LDS address low 32 bits == LDS byte address, so truncate.
__device__ __forceinline__ void asyncCopy16(const float* gaddr, float* lds_generic) {
    uint32_t lds_addr = (uint32_t)(uintptr_t)lds_generic;
    asm volatile("global_load_async_to_lds_b128 %0, %1, off"
                 :: "v"(lds_addr), "v"(gaddr)
                 : "memory");
}

// ---------------- kernel 0: zero the accumulators ----------------
__global__ void cbow_zero_ws(float* ws, int n) {
    int i = blockIdx.x * blockDim.x + threadIdx.x;
    if (i < n) ws[i] = 0.0f;
}

// ---------------- kernel 1: stream U once ----------------
// Fuses:  vi_embed[d] += s[w] * U[w,d]   (GEMV, s[w] = 0.1 * sum_c vi[w,c])
//         neg_embed[m,d] += neg[m,w] * U[w,d]   (WMMA f32 16x16x4 GEMM)
// 512 threads = 16 waves; wave = (m_tile in 0..1, n_tile in 0..7).
__global__ void __launch_bounds__(512)
cbow_stream_U(const float* __restrict__ vi,
              const float* __restrict__ neg,
              const float* __restrict__ U,
              float* __restrict__ ws) {
    __shared__ float U_sh[ROWS * U_STRIDE];        // 33792 B
    __shared__ float A_sh[NEG_PAD * A_STRIDE];     //  8320 B
    __shared__ float s_sh[ROWS];                   //   256 B

    const int tid  = threadIdx.x;
    const int wave = tid >> 5;          // 0..15
    const int lane = tid & 31;

    const int mt   = wave >> 3;         // 0..1  (rows m = 16*mt .. 16*mt+15)
    const int nt   = wave & 7;          // 0..7  (cols n = 16*nt .. 16*nt+15)
    const int mrow = (mt << 4) + (lane & 15);   // A: M = lane%16
    const int ncol = (nt << 4) + (lane & 15);   // B: N = lane%16
    const int kk   = (lane >> 4) << 1;          // lanes 0-15 -> K+{0,1}; 16-31 -> K+{2,3}

    // GEMV ownership: column d, 16-row quarter of each slab
    const int gd = tid & 127;
    const int gq = (tid >> 7) << 4;     // 0,16,32,48

    v8f   c = {};          // 16x16 f32 accumulator, persists across slabs
    float vi_acc = 0.0f;

    for (int slab = blockIdx.x; slab < NSLAB; slab += gridDim.x) {
        const int r0 = slab * ROWS;
        const bool full = (r0 + ROWS) <= VOC;

        // ---- stage U slab (64x128 f32) into padded LDS ----
        if (full) {
            // async global->LDS, 16B per lane, 4 issues per thread
            #pragma unroll
            for (int i = 0; i < 4; ++i) {
                int e = (tid + i * 512) << 2;   // element offset 0..8188
                int w = e >> 7;
                int d = e & 127;
                asyncCopy16(&U[(size_t)(r0 + w) * EDIM + d],
                            &U_sh[w * U_STRIDE + d]);
            }
        } else {
            // tail slab: synchronous, zero-filled
            #pragma unroll
            for (int i = 0; i < 4; ++i) {
                int e = (tid + i * 512) << 2;
                int w = e >> 7;
                int d = e & 127;
                float4 val = make_float4(0.f, 0.f, 0.f, 0.f);
                if (r0 + w < VOC)
                    val = *(const float4*)&U[(size_t)(r0 + w) * EDIM + d];
                *(float4*)&U_sh[w * U_STRIDE + d] = val;
            }
        }
        // ---- stage s[w] = 0.1 * sum_c vi[w,c] ----
        if (tid < ROWS) {
            int w = r0 + tid;
            float sum = 0.0f;
            if (w < VOC) {
                #pragma unroll
                for (int cc = 0; cc < CTX; ++cc)
                    sum += vi[(size_t)w * CTX + cc];
            }
            s_sh[tid] = 0.1f * sum;
        }
        // ---- stage A = neg_samples tile (32x64, rows >=20 zero-padded) ----
        #pragma unroll
        for (int i = 0; i < 4; ++i) {
            int idx = tid + i * 512;            // 0..2047
            int m   = idx >> 6;                 // 0..31
            int k   = idx & 63;
            float v = 0.0f;
            if (m < NNEG && (r0 + k) < VOC)
                v = neg[(size_t)m * VOC + r0 + k];
            A_sh[m * A_STRIDE + k] = v;
        }
        waitAsync0();          // this wave's async group landed in LDS
        __syncthreads();       // all waves' staging visible

        // ---- WMMA: C += A(16x4) x B(4x16), 16 k-steps over the slab ----
        #pragma unroll 4
        for (int k0 = 0; k0 < ROWS; k0 += 4) {
            v2f a, b;
            a.x = A_sh[mrow * A_STRIDE + k0 + kk];
            a.y = A_sh[mrow * A_STRIDE + k0 + kk + 1];
            b.x = U_sh[(k0 + kk)     * U_STRIDE + ncol];
            b.y = U_sh[(k0 + kk + 1) * U_STRIDE + ncol];
            c = __builtin_amdgcn_wmma_f32_16x16x4_f32(
                    /*neg_a=*/false, a, /*neg_b=*/false, b,
                    /*c_mod=*/(short)0, c, /*reuse_a=*/false, /*reuse_b=*/false);
        }

        // ---- fused vi_embed GEMV out of the same LDS slab (all 512 threads) ----
        {
            float g = 0.0f;
            #pragma unroll
            for (int w = gq; w < gq + 16; ++w)
                g += s_sh[w] * U_sh[w * U_STRIDE + gd];
            vi_acc += g;
        }
        __syncthreads();       // LDS reuse next slab
    }

    // ---- write-out: one round of atomics per block ----
    atomAddF(&ws[WS_VI + gd], vi_acc);

    #pragma unroll
    for (int r = 0; r < 8; ++r) {
        // C layout: VGPR r holds M = r + (lane/16)*8, N = lane%16
        int m = (mt << 4) + r + ((lane >> 4) << 3);
        int n = (nt << 4) + (lane & 15);
        atomAddF(&ws[WS_NEG + m * EDIM + n], c[r]);
    }
}

// ---------------- kernel 2: stream V once (vo_embed GEMV) ----------------
__global__ void __launch_bounds__(256)
cbow_stream_V(const float* __restrict__ vo,
              const float* __restrict__ V,
              float* __restrict__ ws) {
    const int tid  = threadIdx.x;       // 256
    const int d    = tid & 127;
    const int half = tid >> 7;          // 0/1: split rows between thread-halves
    float acc = 0.0f;
    for (int slab = blockIdx.x; slab * 128 < VOC; slab += gridDim.x) {
        int r0   = slab * 128;
        int rend = (r0 + 128 < VOC) ? (r0 + 128) : VOC;
        for (int w = r0 + half; w < rend; w += 2)
            acc += vo[w] * V[(size_t)w * EDIM + d];
    }
    atomAddF(&ws[WS_VO + d], acc);
}

// ---------------- kernel 3: scalar finish (one wave32) ----------------
__device__ __forceinline__ float log_sigmoid(float x) {
    return fminf(x, 0.0f) - log1pf(expf(-fabsf(x)));
}

__global__ void cbow_finish(const float* __restrict__ ws, float* __restrict__ out) {
    const int lane = threadIdx.x;       // 32 threads
    // left = log_sigmoid(vi_embed . vo_embed)
    float dot = 0.0f;
    for (int d = lane; d < EDIM; d += 32)
        dot += ws[WS_VO + d] * ws[WS_VI + d];
    for (int off = 16; off > 0; off >>= 1) dot += __shfl_down(dot, off);

    float rsum = 0.0f;
    for (int m = 0; m < NNEG; ++m) {
        float t = 0.0f;
        for (int d = lane; d < EDIM; d += 32)
            t += ws[WS_NEG + m * EDIM + d] * ws[WS_VI + d];
        for (int off = 16; off > 0; off >>= 1) t += __shfl_down(t, off);
        if (lane == 0) rsum += log_sigmoid(-t);
    }
    if (lane == 0)
        out[0] = -(log_sigmoid(dot) + rsum);
}

// ---------------- host launcher ----------------
extern "C" void kernel_launch(void* const* d_in, const int* in_sizes, int n_in,
                              void* d_out, int out_size, void* d_ws, size_t ws_size,
                              hipStream_t stream) {
    const float* vo  = (const float*)d_in[0];
    const float* vi  = (const float*)d_in[1];
    const float* neg = (const float*)d_in[2];
    const float* V   = (const float*)d_in[3];
    const float* U   = (const float*)d_in[4];
    float* out = (float*)d_out;
    float* ws  = (float*)d_ws;

    (void)in_sizes; (void)n_in; (void)out_size; (void)ws_size;

    cbow_zero_ws<<<(WS_N + 255) / 256, 256, 0, stream>>>(ws, WS_N);
    cbow_stream_U<<<390, 512, 0, stream>>>(vi, neg, U, ws);
    cbow_stream_V<<<416, 256, 0, stream>>>(vo, V, ws);
    cbow_finish<<<1, 32, 0, stream>>>(ws, out);
}